// SelFuseFeature_60962765799864
// MI455X (gfx1250) — compile-verified
//
#include <hip/hip_runtime.h>
#include <math.h>

typedef float v2f __attribute__((ext_vector_type(2)));
typedef float v8f __attribute__((ext_vector_type(8)));

#define Nn 8
#define Cc 256
#define Hh 128
#define Ww 128
#define HW (Hh * Ww)            /* 16384 */
#define CHW (Cc * HW)           /* 4194304 elems = 16 MB */
#define XTOT ((size_t)Nn * CHW) /* 33554432 elems = 128 MB */

struct alignas(16) Samp { int r0; int c0; float fr; float fc; };

// ---------------------------------------------------------------------------
// Pass 1: one sweep over x computes, per pixel tile of 16 (consecutive w):
//   N-col 0,1 : df = w_df @ x[c]          (-> out2 = df_raw, + tap table)
//   N-col 2,3 : first half of fuse GEMM   (-> out1, second half added later)
// via V_WMMA_F32_16X16X4_F32 chained over K=256 (64 WMMAs / tile).
// B table in LDS is zero-padded to all 16 N-columns so the inner loop has
// uniform control flow (no exec-mask save/restore around the ds loads).
// ---------------------------------------------------------------------------
__global__ __launch_bounds__(256) void k_df_fuse0(
    const float* __restrict__ x, const float* __restrict__ w_df,
    const float* __restrict__ w_fuse, const float* __restrict__ b_df,
    const float* __restrict__ b_fuse, float* __restrict__ out1,
    float* __restrict__ out2, Samp* __restrict__ samp)
{
    __shared__ float wcol[Cc][16];    // cols 0-3 live, 4-15 zero
    __shared__ float dsh[8][4][16];   // [wave][Ncol][pixel]

    for (int t = threadIdx.x; t < Cc * 16; t += 256) {
        int k = t >> 4, c16 = t & 15;
        float v = 0.0f;
        if (c16 < 2)      v = w_df[c16 * Cc + k];
        else if (c16 < 4) v = w_fuse[(c16 - 2) * (2 * Cc) + k];
        wcol[k][c16] = v;
    }
    __syncthreads();

    int lane = threadIdx.x & 31;
    int wib  = threadIdx.x >> 5;
    int tile = blockIdx.x * 8 + wib;          // 8192 tiles total
    int w0   = (tile & 7) * 16;
    int h    = (tile >> 3) & 127;
    int n    = tile >> 10;
    int m    = lane & 15;                     // A-matrix M index / B-matrix N index
    int sel  = lane >> 4;                     // K sub-select per ISA layout

    const float* xb = x + (size_t)n * CHW + (size_t)h * Ww + w0;

    v8f acc = {};
    for (int k = 0; k < Cc; k += 4) {
        int ka = k + 2 * sel;
        v2f A, B;
        A.x = xb[(size_t)ka * HW + m];        // A(M=m, K=ka)
        A.y = xb[(size_t)(ka + 1) * HW + m];  // A(M=m, K=ka+1)
        B.x = wcol[ka][m];                    // B(K=ka,   N=m)  (zero for m>=4)
        B.y = wcol[ka + 1][m];                // B(K=ka+1, N=m)
        acc = __builtin_amdgcn_wmma_f32_16x16x4_f32(
            false, A, false, B, (short)0, acc, false, false);
    }

    // D layout: VGPR q, lanes 0-15 -> (M=q, N=lane); lanes 16-31 -> (M=q+8, N=lane-16)
    if (m < 4) {
        #pragma unroll
        for (int q = 0; q < 8; ++q) dsh[wib][m][sel * 8 + q] = acc[q];
    }
    __syncthreads();

    if (lane < 16) {
        int p = w0 + lane;
        float df0 = dsh[wib][0][lane] + b_df[0];
        float df1 = dsh[wib][1][lane] + b_df[1];
        float f0  = dsh[wib][2][lane] + b_fuse[0];
        float f1  = dsh[wib][3][lane] + b_fuse[1];

        size_t o = (size_t)n * (2 * HW) + (size_t)h * Ww + p;
        out1[o]      = f0;   out1[o + HW] = f1;   // partial fuse (x half)
        out2[o]      = df0;  out2[o + HW] = df1;  // df_raw

        // mask: zero displacement unless |df| > 0.5
        float m2 = df0 * df0 + df1 * df1;
        float d0 = (m2 > 0.25f) ? df0 : 0.0f;
        float d1 = (m2 > 0.25f) ? df1 : 0.0f;

        // df at (h,w=p) drives OUTPUT pixel (i=p, j=h):
        //   row = i + df1[n,h,p],  col = j + df0[n,h,p]   (border clip, align_corners)
        float rowf = fminf(fmaxf((float)p + d1, 0.0f), 127.0f);
        float colf = fminf(fmaxf((float)h + d0, 0.0f), 127.0f);
        float r0f = floorf(rowf), c0f = floorf(colf);
        Samp s;
        s.r0 = (int)r0f;  s.c0 = (int)c0f;
        s.fr = rowf - r0f; s.fc = colf - c0f;
        samp[(size_t)n * HW + (size_t)p * Ww + h] = s;
    }
}

// ---------------------------------------------------------------------------
// Pass 2 (x20): bilinear gather with the precomputed, channel-invariant taps.
// 32 consecutive j per wave (coalesced), 32 channels per thread.
// sx (128 MB) fits in the 192 MB L2, so the 4 taps are L2 hits; HBM sees
// ~1 read + 1 write of 128 MB per iteration.
// ---------------------------------------------------------------------------
__global__ __launch_bounds__(256) void k_sample(
    const float* __restrict__ in, const Samp* __restrict__ samp,
    float* __restrict__ outp)
{
    int gt = blockIdx.x * 256 + threadIdx.x;  // 2^20 threads
    int j  = gt & 127;
    int i  = (gt >> 7) & 127;
    int cg = (gt >> 14) & 7;                  // channel group of 32
    int n  = gt >> 17;

    Samp s = samp[(size_t)n * HW + (size_t)i * Ww + j];
    int r1 = min(s.r0 + 1, 127);
    int c1 = min(s.c0 + 1, 127);
    int o00 = s.r0 * Ww + s.c0;
    int o01 = s.r0 * Ww + c1;
    int o10 = r1 * Ww + s.c0;
    int o11 = r1 * Ww + c1;
    float w11 = s.fr * s.fc;
    float w10 = s.fr - w11;                   // fr*(1-fc)
    float w01 = s.fc - w11;                   // (1-fr)*fc
    float w00 = 1.0f - s.fr - s.fc + w11;     // (1-fr)*(1-fc)

    const float* ip = in   + (size_t)n * CHW + (size_t)cg * (32 * HW);
    float*       op = outp + (size_t)n * CHW + (size_t)cg * (32 * HW)
                           + (size_t)i * Ww + j;
    #pragma unroll 4
    for (int c = 0; c < 32; ++c) {
        const float* pl = ip + (size_t)c * HW;
        float v = fmaf(w00, pl[o00],
                  fmaf(w01, pl[o01],
                  fmaf(w10, pl[o10], w11 * pl[o11])));
        op[(size_t)c * HW] = v;
    }
}

// ---------------------------------------------------------------------------
// Pass 3: second half of fuse GEMM over final sx, accumulated into out1.
// Same WMMA tiling as pass 1; B has 2 live columns (w_fuse[:,256:512]),
// zero-padded to 16 for uniform control flow.
// ---------------------------------------------------------------------------
__global__ __launch_bounds__(256) void k_fuse1(
    const float* __restrict__ sx, const float* __restrict__ w_fuse,
    float* __restrict__ out1)
{
    __shared__ float wcol[Cc][16];    // cols 0-1 live, 2-15 zero
    __shared__ float dsh[8][2][16];

    for (int t = threadIdx.x; t < Cc * 16; t += 256) {
        int k = t >> 4, c16 = t & 15;
        wcol[k][c16] = (c16 < 2) ? w_fuse[c16 * (2 * Cc) + Cc + k] : 0.0f;
    }
    __syncthreads();

    int lane = threadIdx.x & 31;
    int wib  = threadIdx.x >> 5;
    int tile = blockIdx.x * 8 + wib;
    int w0   = (tile & 7) * 16;
    int h    = (tile >> 3) & 127;
    int n    = tile >> 10;
    int m    = lane & 15;
    int sel  = lane >> 4;

    const float* xb = sx + (size_t)n * CHW + (size_t)h * Ww + w0;

    v8f acc = {};
    for (int k = 0; k < Cc; k += 4) {
        int ka = k + 2 * sel;
        v2f A, B;
        A.x = xb[(size_t)ka * HW + m];
        A.y = xb[(size_t)(ka + 1) * HW + m];
        B.x = wcol[ka][m];
        B.y = wcol[ka + 1][m];
        acc = __builtin_amdgcn_wmma_f32_16x16x4_f32(
            false, A, false, B, (short)0, acc, false, false);
    }

    if (m < 2) {
        #pragma unroll
        for (int q = 0; q < 8; ++q) dsh[wib][m][sel * 8 + q] = acc[q];
    }
    __syncthreads();

    if (lane < 16) {
        int p = w0 + lane;
        size_t o = (size_t)n * (2 * HW) + (size_t)h * Ww + p;
        out1[o]      += dsh[wib][0][lane];
        out1[o + HW] += dsh[wib][1][lane];
    }
}

// ---------------------------------------------------------------------------
extern "C" void kernel_launch(void* const* d_in, const int* in_sizes, int n_in,
                              void* d_out, int out_size, void* d_ws, size_t ws_size,
                              hipStream_t stream) {
    (void)in_sizes; (void)n_in; (void)out_size; (void)ws_size;
    const float* x      = (const float*)d_in[0];
    const float* w_df   = (const float*)d_in[1];
    const float* b_df   = (const float*)d_in[2];
    const float* w_fuse = (const float*)d_in[3];
    const float* b_fuse = (const float*)d_in[4];

    float* out1 = (float*)d_out;                   // fused output (N,2,H,W)
    float* out2 = out1 + (size_t)Nn * 2 * HW;      // df_raw       (N,2,H,W)

    float* bufA = (float*)d_ws;                    // 128 MB
    float* bufB = bufA + XTOT;                     // 128 MB
    Samp*  samp = (Samp*)(bufB + XTOT);            // 2 MB tap table

    // Pass 1: df + tap table + fuse(x-half), single sweep over x.
    k_df_fuse0<<<1024, 256, 0, stream>>>(x, w_df, w_fuse, b_df, b_fuse,
                                         out1, out2, samp);

    // Pass 2: 20 iterated bilinear samples, ping-pong A/B.
    const float* sin = x;
    for (int it = 0; it < 20; ++it) {
        float* sout = (it & 1) ? bufB : bufA;
        k_sample<<<4096, 256, 0, stream>>>(sin, samp, sout);
        sin = sout;
    }
    // sin == bufB after 20 iterations.

    // Pass 3: fuse(sx-half) accumulated into out1.
    k_fuse1<<<1024, 256, 0, stream>>>(sin, w_fuse, out1);
}